// ContrastiveLoss_71494025609465
// MI455X (gfx1250) — compile-verified
//
#include <hip/hip_runtime.h>
#include <math.h>

// ---------------------------------------------------------------------------
// SCAN t2i contrastive loss for MI455X (gfx1250, wave32, WMMA + TDM).
// One workgroup (128 thr = 4 waves) per (caption c, image i) pair.
//   Pass 1: attn(36x50) = im(36x1024) * s(50x1024)^T via V_WMMA_F32_16X16X4_F32,
//           k-chunks staged by tensor_load_to_lds (TDM DMA, double-buffered,
//           hardware pad 4 DW / 64 DW reproduces the bank-conflict-free ST=68).
//   Epilogue: leaky_relu -> word mask -> row L2-norm -> softmax over regions.
//   Pass 2: wei tiles = attn_t(50x48pad) * im via WMMA, folded immediately into
//           w12 / |s|^2 / |wei|^2 (wei never materialized), LSE -> scores[i][c].
// Kernel 2: hinge loss with hardest negatives over the 128x128 score matrix.
// ---------------------------------------------------------------------------

typedef float v2f __attribute__((ext_vector_type(2)));
typedef float v4f __attribute__((ext_vector_type(4)));
typedef float v8f __attribute__((ext_vector_type(8)));
typedef unsigned int v4u __attribute__((ext_vector_type(4)));
typedef int v4i __attribute__((ext_vector_type(4)));
typedef int v8i __attribute__((ext_vector_type(8)));

#ifndef __has_builtin
#define __has_builtin(x) 0
#endif
#if defined(__HIP_DEVICE_COMPILE__) &&                      \
    __has_builtin(__builtin_amdgcn_tensor_load_to_lds) &&   \
    __has_builtin(__builtin_amdgcn_s_wait_tensorcnt)
#define USE_TDM 1
#endif

#define NI 128
#define NR 36      // regions
#define NW 50      // words
#define ND 1024    // feature dim
#define NCH (ND / KC)
#define KC 64      // k-chunk staged in LDS
#define ST 68      // im/s LDS row stride (floats): 272B -> 4-bank step/row
#define ATS 52     // attn_t LDS row stride (floats): 208B, conflict-free
#define RP 48      // padded regions (3 x 16)
#define WP 64      // padded words   (4 x 16)

// LDS pool layout (floats). attn aliases IM1 (dead after pass 1 compute);
// attn_t aliases S0 (s buffers unused in pass 2).
#define IM0_OFF 0
#define IM1_OFF (RP * ST)                  // 3264
#define S0_OFF (2 * RP * ST)               // 6528
#define S1_OFF (2 * RP * ST + WP * ST)     // 10880
#define POOL_SZ (2 * RP * ST + 2 * WP * ST)  // 15232 floats ~ 59.5 KB
#define ATTN_OFF IM1_OFF
#define AT_OFF S0_OFF                      // WP*ATS = 3328 <= WP*ST = 4352

#define LAM_SM 9.0f
#define LAM_LSE 6.0f
#define MARGIN 0.2f
#define EPSV 1e-8f

#ifdef USE_TDM
// 2D tile DMA: rows x 64 fp32 elements, row stride 1024 elements, with
// hardware LDS padding of 4 DWORDs after every 64 DWORDs (-> ST = 68).
// This toolchain exposes the 6-argument builtin (4 D# groups + extra group +
// cpol); unused trailing groups are zero-filled.
__device__ __forceinline__ void tdm_load_tile(unsigned lds_byte_off,
                                              const float* gsrc, int rows) {
  unsigned long long ga = (unsigned long long)(uintptr_t)gsrc;
  v4u g0 = {
      1u,                                          // count=1 (valid user D#)
      lds_byte_off,                                // lds_addr (bytes)
      (unsigned)(ga & 0xffffffffu),                // global_addr[31:0]
      (unsigned)((ga >> 32) & 0x01ffffffu) | (2u << 30)  // ga[56:32], type=2
  };
  v8i g1 = {
      (int)((2u << 16) | (1u << 20) | (5u << 22) | (3u << 25)),
      // data_size=4B | pad_enable | pad_interval: 64 DW | pad_amount: 4 DW
      (int)((unsigned)ND << 16),                   // tensor_dim0 = 1024
      (int)((unsigned)rows << 16),                 // tensor_dim1 = rows
      (int)((unsigned)KC << 16),                   // tile_dim0 = 64
      rows,                                        // tile_dim1 = rows
      (int)ND,                                     // tensor_dim0_stride = 1024
      0, 0};
  v4i gz4 = {0, 0, 0, 0};
  v8i gz8 = {0, 0, 0, 0, 0, 0, 0, 0};
  __builtin_amdgcn_tensor_load_to_lds(g0, g1, gz4, gz4, gz8, 0);
}
#endif

__device__ __forceinline__ unsigned lds_off_bytes(const float* p) {
  return (unsigned)(unsigned long long)(uintptr_t)p;
}

__launch_bounds__(128)
__global__ void scan_scores_kernel(const float* __restrict__ im,
                                   const float* __restrict__ s,
                                   const int* __restrict__ s_l,
                                   float* __restrict__ scores) {
  __shared__ __align__(16) float pool[POOL_SZ];
  __shared__ float rinv[RP];
  __shared__ float w12s[WP];
  __shared__ float w1s[WP];
  __shared__ float w2s[WP];
  __shared__ float es[WP];

  const int c = blockIdx.x;  // caption
  const int i = blockIdx.y;  // image
  const int tid = threadIdx.x;
  const int wv = tid >> 5;
  const int lane = tid & 31;
  const int lhalf = lane >> 4;
  const int l16 = lane & 15;
  const int s_len = s_l[c];

  const float* __restrict__ imB = im + (size_t)i * NR * ND;
  const float* __restrict__ sB = s + (size_t)c * NW * ND;

  // Zero padded rows of all 4 staging buffers (TDM never touches them) and
  // the reduction arrays.
  {
    const v4f z = {0.f, 0.f, 0.f, 0.f};
    for (int idx = tid; idx < 2 * (RP - NR) * (KC / 4); idx += 128) {
      int b = idx >= (RP - NR) * (KC / 4);
      int r = idx - b * (RP - NR) * (KC / 4);
      int row = NR + (r >> 4), c4 = r & 15;
      *(v4f*)&pool[(b ? IM1_OFF : IM0_OFF) + row * ST + c4 * 4] = z;
    }
    for (int idx = tid; idx < 2 * (WP - NW) * (KC / 4); idx += 128) {
      int b = idx >= (WP - NW) * (KC / 4);
      int r = idx - b * (WP - NW) * (KC / 4);
      int row = NW + (r >> 4), c4 = r & 15;
      *(v4f*)&pool[(b ? S1_OFF : S0_OFF) + row * ST + c4 * 4] = z;
    }
    if (tid < WP) {
      w12s[tid] = 0.f;
      w1s[tid] = 0.f;
      w2s[tid] = 0.f;
    }
  }

  const unsigned poolB = lds_off_bytes(pool);

  // ---------------- Pass 1: attn = im * s^T ----------------
#ifdef USE_TDM
  if (wv == 0) {  // prologue DMA: chunk 0 -> buffer 0
    tdm_load_tile(poolB + IM0_OFF * 4, imB, NR);
    tdm_load_tile(poolB + S0_OFF * 4, sB, NW);
  }
#endif

  v8f acc0 = {0.f, 0.f, 0.f, 0.f, 0.f, 0.f, 0.f, 0.f};
  v8f acc1 = acc0;
  v8f acc2 = acc0;

  for (int kc = 0; kc < NCH; ++kc) {
    const int k0 = kc * KC;
    __syncthreads();  // chunk kc-1 consumers done -> safe to refill its buffer
#ifdef USE_TDM
    if (wv == 0) {
      if (kc + 1 < NCH) {  // prefetch chunk kc+1 into the other buffer
        unsigned nb = (unsigned)((kc + 1) & 1);
        tdm_load_tile(poolB + (nb ? IM1_OFF : IM0_OFF) * 4, imB + k0 + KC, NR);
        tdm_load_tile(poolB + (nb ? S1_OFF : S0_OFF) * 4, sB + k0 + KC, NW);
        __builtin_amdgcn_s_wait_tensorcnt(2);  // chunk kc's 2 DMAs complete
      } else {
        __builtin_amdgcn_s_wait_tensorcnt(0);
      }
    }
#else
    {
      float* imD = pool + ((kc & 1) ? IM1_OFF : IM0_OFF);
      float* sD = pool + ((kc & 1) ? S1_OFF : S0_OFF);
      for (int idx = tid; idx < NR * (KC / 4); idx += 128) {
        int row = idx >> 4, c4 = idx & 15;
        *(v4f*)&imD[row * ST + c4 * 4] =
            *(const v4f*)(imB + (size_t)row * ND + k0 + c4 * 4);
      }
      for (int idx = tid; idx < NW * (KC / 4); idx += 128) {
        int row = idx >> 4, c4 = idx & 15;
        *(v4f*)&sD[row * ST + c4 * 4] =
            *(const v4f*)(sB + (size_t)row * ND + k0 + c4 * 4);
      }
    }
#endif
    __syncthreads();

    const float* imL = pool + ((kc & 1) ? IM1_OFF : IM0_OFF);
    const float* sL = pool + ((kc & 1) ? S1_OFF : S0_OFF);
    const int koff = 2 * lhalf;  // K split across lane halves (16x4 f32 frag)
#pragma unroll 4
    for (int kk = 0; kk < KC; kk += 4) {
      v2f b = *(const v2f*)&sL[(wv * 16 + l16) * ST + kk + koff];
      v2f a0 = *(const v2f*)&imL[(0 + l16) * ST + kk + koff];
      v2f a1 = *(const v2f*)&imL[(16 + l16) * ST + kk + koff];
      v2f a2 = *(const v2f*)&imL[(32 + l16) * ST + kk + koff];
      acc0 = __builtin_amdgcn_wmma_f32_16x16x4_f32(false, a0, false, b,
                                                   (short)0, acc0, false, false);
      acc1 = __builtin_amdgcn_wmma_f32_16x16x4_f32(false, a1, false, b,
                                                   (short)0, acc1, false, false);
      acc2 = __builtin_amdgcn_wmma_f32_16x16x4_f32(false, a2, false, b,
                                                   (short)0, acc2, false, false);
    }
  }
  __syncthreads();  // pass-1 reads done: attn may overwrite im buffer 1

  // Epilogue 1a: leaky_relu(0.1) + word mask, spill attn tiles to LDS.
  {
    float* attnL = pool + ATTN_OFF;
    const int w = wv * 16 + l16;
    const float keep = (w < s_len) ? 1.f : 0.f;
#pragma unroll
    for (int v = 0; v < 8; ++v) {
      const int r0 = v + 8 * lhalf;  // C/D: VGPR v -> row v (v+8 hi lanes)
      float x0 = acc0[v];
      float x1 = acc1[v];
      float x2 = acc2[v];
      x0 = (x0 > 0.f ? x0 : 0.1f * x0) * keep;
      x1 = (x1 > 0.f ? x1 : 0.1f * x1) * keep;
      x2 = (x2 > 0.f ? x2 : 0.1f * x2) * keep;
      attnL[(0 + r0) * ST + w] = x0;
      attnL[(16 + r0) * ST + w] = x1;
      attnL[(32 + r0) * ST + w] = x2;
    }
  }
  __syncthreads();

  // Epilogue 1b: L2 norm over words per region row.
  if (tid < NR) {
    const float* attnL = pool + ATTN_OFF;
    float sum = 0.f;
#pragma unroll 8
    for (int w = 0; w < WP; ++w) {
      float x = attnL[tid * ST + w];
      sum += x * x;
    }
    rinv[tid] = 1.f / (sqrtf(sum) + EPSV);
  }
  __syncthreads();

  // Epilogue 1c: softmax over regions (scale 9) -> attn_t[w][r].
  if (tid < WP) {
    const float* attnL = pool + ATTN_OFF;
    float* atL = pool + AT_OFF;
    const int w = tid;
    float mx = -1e30f;
    for (int r = 0; r < NR; ++r) {
      float v = attnL[r * ST + w] * rinv[r] * LAM_SM;
      mx = fmaxf(mx, v);
    }
    float den = 0.f;
    for (int r = 0; r < NR; ++r) {
      float e = __expf(attnL[r * ST + w] * rinv[r] * LAM_SM - mx);
      atL[w * ATS + r] = e;
      den += e;
    }
    const float id = 1.f / den;
    for (int r = 0; r < NR; ++r) atL[w * ATS + r] *= id;
    for (int r = NR; r < RP; ++r) atL[w * ATS + r] = 0.f;  // exact K padding
  }
  __syncthreads();

  // ---------------- Pass 2: wei = attn_t * im, fold into w12/w1/w2 ---------
  float w12a[8], w1a[8], w2a[8];
#pragma unroll
  for (int v = 0; v < 8; ++v) w12a[v] = w1a[v] = w2a[v] = 0.f;

#ifdef USE_TDM
  if (wv == 0) tdm_load_tile(poolB + IM0_OFF * 4, imB, NR);
#endif

  for (int kc = 0; kc < NCH; ++kc) {
    const int k0 = kc * KC;
    __syncthreads();
#ifdef USE_TDM
    if (wv == 0) {
      if (kc + 1 < NCH) {
        unsigned nb = (unsigned)((kc + 1) & 1);
        tdm_load_tile(poolB + (nb ? IM1_OFF : IM0_OFF) * 4, imB + k0 + KC, NR);
        __builtin_amdgcn_s_wait_tensorcnt(1);
      } else {
        __builtin_amdgcn_s_wait_tensorcnt(0);
      }
    }
#else
    {
      float* imD = pool + ((kc & 1) ? IM1_OFF : IM0_OFF);
      for (int idx = tid; idx < NR * (KC / 4); idx += 128) {
        int row = idx >> 4, c4 = idx & 15;
        *(v4f*)&imD[row * ST + c4 * 4] =
            *(const v4f*)(imB + (size_t)row * ND + k0 + c4 * 4);
      }
    }
#endif
    __syncthreads();

    const float* imL = pool + ((kc & 1) ? IM1_OFF : IM0_OFF);
    const float* atL = pool + AT_OFF;
    const int koff = 2 * lhalf;
#pragma unroll
    for (int t = 0; t < 4; ++t) {  // 16-wide d sub-tiles of this chunk
      const int dcol = t * 16 + l16;
      v8f acc = {0.f, 0.f, 0.f, 0.f, 0.f, 0.f, 0.f, 0.f};
#pragma unroll
      for (int kk = 0; kk < RP; kk += 4) {
        v2f a = *(const v2f*)&atL[(wv * 16 + l16) * ATS + kk + koff];
        const int r0 = kk + koff;
        v2f b = {imL[r0 * ST + dcol], imL[(r0 + 1) * ST + dcol]};
        acc = __builtin_amdgcn_wmma_f32_16x16x4_f32(false, a, false, b,
                                                    (short)0, acc, false, false);
      }
#pragma unroll
      for (int v = 0; v < 8; ++v) {
        const int w = wv * 16 + v + 8 * lhalf;
        // s is L2-resident; 64B coalesced per half-wave; guard padded words.
        float sval = (w < NW) ? sB[(size_t)w * ND + k0 + dcol] : 0.f;
        float we = acc[v];
        w12a[v] += we * sval;
        w2a[v] += we * we;
        w1a[v] += sval * sval;
      }
    }
  }

#pragma unroll
  for (int v = 0; v < 8; ++v) {
    const int w = wv * 16 + v + 8 * lhalf;
    atomicAdd(&w12s[w], w12a[v]);  // ds_add_f32
    atomicAdd(&w2s[w], w2a[v]);
    atomicAdd(&w1s[w], w1a[v]);
  }
  __syncthreads();

  // cosine similarity + masked exp(6*sim)
  if (tid < WP) {
    float w1v = sqrtf(w1s[tid]);
    float w2v = sqrtf(w2s[tid]);
    float den = fmaxf(w1v * w2v, EPSV);
    float rs = w12s[tid] / den;
    es[tid] = (tid < s_len) ? __expf(LAM_LSE * rs) : 0.f;
  }
  __syncthreads();

  if (tid == 0) {
    float sum = 0.f;
    for (int w = 0; w < WP; ++w) sum += es[w];
    scores[(size_t)i * NI + c] = __logf(sum) / LAM_LSE;  // scores[i][c]
  }
}

__launch_bounds__(128)
__global__ void scan_loss_kernel(const float* __restrict__ scores,
                                 float* __restrict__ out) {
  __shared__ float red[128];
  const int t = threadIdx.x;
  const float dt = scores[(size_t)t * NI + t];
  float m1 = 0.f;  // hardest negative in row t   (cost_s)
  float m2 = 0.f;  // hardest negative in col t   (cost_im)
  for (int j = 0; j < NI; ++j) {
    if (j != t) {
      float v1 = MARGIN + scores[(size_t)t * NI + j] - dt;
      m1 = fmaxf(m1, fmaxf(v1, 0.f));
      float v2 = MARGIN + scores[(size_t)j * NI + t] - dt;
      m2 = fmaxf(m2, fmaxf(v2, 0.f));
    }
  }
  red[t] = m1 + m2;
  __syncthreads();
  for (int off = 64; off > 0; off >>= 1) {
    if (t < off) red[t] += red[t + off];
    __syncthreads();
  }
  if (t == 0) out[0] = red[0];
}

extern "C" void kernel_launch(void* const* d_in, const int* in_sizes, int n_in,
                              void* d_out, int out_size, void* d_ws,
                              size_t ws_size, hipStream_t stream) {
  (void)in_sizes;
  (void)n_in;
  (void)out_size;
  (void)ws_size;
  const float* im = (const float*)d_in[0];
  const float* s = (const float*)d_in[1];
  const int* s_l = (const int*)d_in[2];
  float* out = (float*)d_out;
  float* scores = (float*)d_ws;  // 128*128 floats = 64 KB scratch

  dim3 grid(NI, NI);
  scan_scores_kernel<<<grid, 128, 0, stream>>>(im, s, s_l, scores);
  scan_loss_kernel<<<1, 128, 0, stream>>>(scores, out);
}